// SSIMLoss_31044023615981
// MI455X (gfx1250) — compile-verified
//
#include <hip/hip_runtime.h>
#include <math.h>

// ---------------- problem geometry ----------------
#define IMG_H 512
#define IMG_W 512
#define NPLANES 48                 // B*C = 16*3 depthwise planes
#define TILES_X (IMG_W / 16)       // 32
#define TILES_Y (IMG_H / 16)       // 32
#define TILES_PER_PLANE (TILES_X * TILES_Y)      // 1024
#define NTILES (NPLANES * TILES_PER_PLANE)       // 49152
#define WAVES_PER_BLOCK 8
#define BLOCK_THREADS (WAVES_PER_BLOCK * 32)     // 256
#define NBLOCKS (NTILES / WAVES_PER_BLOCK)       // 6144
#define LDS_ROW_STRIDE 20          // floats (bank-conflict padding)
#define LDS_ROWS 32                // rows 0..27 live, 28..31 dead scratch

#define C1_CONST 0.0001f
#define C2_CONST 0.0009f

typedef float v2f __attribute__((ext_vector_type(2)));
typedef float v8f __attribute__((ext_vector_type(8)));
typedef float v2f_u __attribute__((ext_vector_type(2), aligned(4)));  // align-4 packed pair

// 11-tap gaussian (sigma=1.5, normalized), zero-padded so that
// GWPAD[d + 15] == g[d] for d in [0,10], 0 otherwise, for d in [-15, 27].
__device__ __constant__ float GWPAD[43] = {
    0.f, 0.f, 0.f, 0.f, 0.f, 0.f, 0.f, 0.f, 0.f, 0.f, 0.f, 0.f, 0.f, 0.f, 0.f,
    0.00102838f, 0.00759875f, 0.03600077f, 0.10936069f, 0.21300553f,
    0.26601172f,
    0.21300553f, 0.10936069f, 0.03600077f, 0.00759875f, 0.00102838f,
    0.f, 0.f, 0.f, 0.f, 0.f, 0.f, 0.f, 0.f, 0.f, 0.f, 0.f, 0.f, 0.f, 0.f, 0.f,
    0.f, 0.f };

__global__ __launch_bounds__(BLOCK_THREADS)
void ssim_tiles_kernel(const float* __restrict__ x,
                       const float* __restrict__ y,
                       float* __restrict__ partial) {
    __shared__ float tbuf[WAVES_PER_BLOCK][LDS_ROWS * LDS_ROW_STRIDE];
    __shared__ float red[BLOCK_THREADS];

    const int lane  = threadIdx.x & 31;
    const int wv    = threadIdx.x >> 5;
    const int col   = lane & 15;               // N index (and tile row for A frags)
    const int khalf = (lane >> 4) ? 2 : 0;     // K offset for upper lane half

    const int tile  = blockIdx.x * WAVES_PER_BLOCK + wv;
    const int plane = tile / TILES_PER_PLANE;
    const int trem  = tile % TILES_PER_PLANE;
    const int ty0   = (trem / TILES_X) * 16;
    const int tx0   = (trem % TILES_X) * 16;

    const float* xp = x + (size_t)plane * (IMG_H * IMG_W);
    const float* yp = y + (size_t)plane * (IMG_H * IMG_W);

    // Constant Gaussian fragments: serve as B (Gc, 28x16) in pass 1 and as
    // A (Gr, 16x28) in pass 2; both are g[K - col] with K = 4c + v + khalf.
    v2f gfrag[7];
#pragma unroll
    for (int c = 0; c < 7; ++c) {
        v2f_u g = *(const v2f_u*)&GWPAD[4 * c + khalf + 15 - col];
        gfrag[c].x = g.x;
        gfrag[c].y = g.y;
    }

    // --- load 32x28 input region (zero-padded) in A-fragment layout ---
    // fragment [h][c]: A row M = col (region row 16h+col), K pair at 4c+khalf
    v2f xa[2][7], ya[2][7];
    const bool interior = (ty0 >= 5) && (ty0 + 26 < IMG_H) &&
                          (tx0 >= 5) && (tx0 + 22 < IMG_W);
    if (interior) {
        // wave-uniform fast path: no bounds checks, packed 8B loads
#pragma unroll
        for (int h = 0; h < 2; ++h) {
            const size_t base =
                (size_t)(ty0 - 5 + 16 * h + col) * IMG_W + (tx0 - 5 + khalf);
            const float* xr = xp + base;
            const float* yr = yp + base;
#pragma unroll
            for (int c = 0; c < 7; ++c) {
                v2f_u tx = *(const v2f_u*)(xr + 4 * c);
                v2f_u ty = *(const v2f_u*)(yr + 4 * c);
                xa[h][c].x = tx.x; xa[h][c].y = tx.y;
                ya[h][c].x = ty.x; ya[h][c].y = ty.y;
            }
        }
    } else {
        // edge tiles: per-element zero-padding guards
#pragma unroll
        for (int h = 0; h < 2; ++h) {
#pragma unroll
            for (int c = 0; c < 7; ++c) {
                int r  = ty0 - 5 + 16 * h + col;     // image row
                int k0 = tx0 - 5 + 4 * c + khalf;    // image col of element 0
                bool rok = (r >= 0) & (r < IMG_H);
                float vx0 = 0.0f, vy0 = 0.0f, vx1 = 0.0f, vy1 = 0.0f;
                if (rok && k0 >= 0 && k0 < IMG_W) {
                    size_t off = (size_t)r * IMG_W + k0;
                    vx0 = xp[off]; vy0 = yp[off];
                }
                if (rok && (k0 + 1) >= 0 && (k0 + 1) < IMG_W) {
                    size_t off = (size_t)r * IMG_W + (k0 + 1);
                    vx1 = xp[off]; vy1 = yp[off];
                }
                xa[h][c].x = vx0; xa[h][c].y = vx1;
                ya[h][c].x = vy0; ya[h][c].y = vy1;
            }
        }
    }

    // precomputed LDS addresses (pass-1 spill base row and pass-2 read base)
    const int spill_row0 = (lane < 16) ? 0 : 8;       // + v + 16h
    float* lds = &tbuf[wv][0];

    // --- 5 fused depthwise convs: m = x, y, x*x, y*y, x*y ---
    v8f res[5];
#pragma unroll
    for (int m = 0; m < 5; ++m) {
        // pass 1 (horizontal): two 16x16 D tiles covering region rows 0..31
#pragma unroll
        for (int h = 0; h < 2; ++h) {
            v8f d = {0.f, 0.f, 0.f, 0.f, 0.f, 0.f, 0.f, 0.f};
#pragma unroll
            for (int c = 0; c < 7; ++c) {
                v2f a;
                if      (m == 0) a = xa[h][c];
                else if (m == 1) a = ya[h][c];
                else if (m == 2) a = xa[h][c] * xa[h][c];
                else if (m == 3) a = ya[h][c] * ya[h][c];
                else             a = xa[h][c] * ya[h][c];
                d = __builtin_amdgcn_wmma_f32_16x16x4_f32(
                        false, a, false, gfrag[c], (short)0, d, false, false);
            }
            // spill D tile to LDS row-major; rows 28..31 are dead scratch,
            // so every store is unconditional (no EXEC juggling)
#pragma unroll
            for (int v = 0; v < 8; ++v) {
                int R = 16 * h + spill_row0 + v;              // 0..31
                lds[R * LDS_ROW_STRIDE + col] = d[v];
            }
        }

        // pass 2 (vertical): out = Gr(16x28) x tmp(28x16); rows 26/27 have g=0.
        // Preload ALL b-fragments first so the WMMA chain isn't serialized on
        // per-chunk ds_load -> s_wait_dscnt 0 round trips (DS returns in order).
        v2f bfr[7];
#pragma unroll
        for (int c = 0; c < 7; ++c) {
            int kb = 4 * c + khalf;
            bfr[c].x = lds[(kb + 0) * LDS_ROW_STRIDE + col];
            bfr[c].y = lds[(kb + 1) * LDS_ROW_STRIDE + col];
        }
        v8f o = {0.f, 0.f, 0.f, 0.f, 0.f, 0.f, 0.f, 0.f};
#pragma unroll
        for (int c = 0; c < 7; ++c) {
            o = __builtin_amdgcn_wmma_f32_16x16x4_f32(
                    false, gfrag[c], false, bfr[c], (short)0, o, false, false);
        }
        res[m] = o;
    }

    // --- SSIM map + per-thread partial sum (8 pixels/lane) ---
    float s = 0.0f;
#pragma unroll
    for (int e = 0; e < 8; ++e) {
        float mx  = res[0][e];
        float my  = res[1][e];
        float sxx = res[2][e] - mx * mx;
        float syy = res[3][e] - my * my;
        float sxy = res[4][e] - mx * my;
        float num = (2.0f * mx * my + C1_CONST) * (2.0f * sxy + C2_CONST);
        float den = (mx * mx + my * my + C1_CONST) * (sxx + syy + C2_CONST);
        s += num / den;
    }

    // --- deterministic block reduction ---
    red[threadIdx.x] = s;
    __syncthreads();
#pragma unroll
    for (int st = BLOCK_THREADS / 2; st > 0; st >>= 1) {
        if (threadIdx.x < st) red[threadIdx.x] += red[threadIdx.x + st];
        __syncthreads();
    }
    if (threadIdx.x == 0) partial[blockIdx.x] = red[0];
}

__global__ __launch_bounds__(256)
void ssim_finalize_kernel(const float* __restrict__ partial, int n,
                          float* __restrict__ out) {
    __shared__ float red[256];
    float s = 0.0f;
    for (int i = threadIdx.x; i < n; i += 256) s += partial[i];   // fixed order
    red[threadIdx.x] = s;
    __syncthreads();
#pragma unroll
    for (int st = 128; st > 0; st >>= 1) {
        if (threadIdx.x < st) red[threadIdx.x] += red[threadIdx.x + st];
        __syncthreads();
    }
    if (threadIdx.x == 0) {
        float mean = red[0] / (float)((size_t)NPLANES * IMG_H * IMG_W);
        out[0] = -log10f(mean);   // loss
        out[1] = mean;            // ssim_mean
    }
}

extern "C" void kernel_launch(void* const* d_in, const int* in_sizes, int n_in,
                              void* d_out, int out_size, void* d_ws, size_t ws_size,
                              hipStream_t stream) {
    const float* x = (const float*)d_in[0];
    const float* y = (const float*)d_in[1];
    float* out     = (float*)d_out;
    float* partial = (float*)d_ws;   // NBLOCKS floats, all written by stage 1

    ssim_tiles_kernel<<<dim3(NBLOCKS), dim3(BLOCK_THREADS), 0, stream>>>(x, y, partial);
    ssim_finalize_kernel<<<dim3(1), dim3(256), 0, stream>>>(partial, NBLOCKS, out);
}